// fully_fix_linear_23098334117999
// MI455X (gfx1250) — compile-verified
//
#include <hip/hip_runtime.h>
#include <math.h>

#ifndef __has_builtin
#define __has_builtin(x) 0
#endif

#if __has_builtin(__builtin_amdgcn_global_load_async_to_lds_b128)
#define USE_ASYNC_LDS 1
#else
#define USE_ASYNC_LDS 0
#endif

typedef int v4i __attribute__((ext_vector_type(4)));
typedef __attribute__((address_space(1))) v4i* gptr_v4i;   // global
typedef __attribute__((address_space(3))) v4i* lptr_v4i;   // LDS

static constexpr int kWavesPerBlock = 8;
static constexpr int kIn        = 1024;               // IN dimension (reference-fixed)
static constexpr int kSegFloats = kIn / 32;           // 32 floats per lane
static constexpr int kSegBytes  = kSegFloats * 4 + 16;// 144 B: pad breaks LDS bank conflicts
static constexpr int kRowLds    = 32 * kSegBytes;     // 4608 B per row

// Saturating-accumulate monoid element: f(a) = min(max(a + S, L), H).
// compose(outer, inner)(a) = outer(inner(a)):
//   S = Si + So;  L = max(Li + So, Lo);  H = min(max(Hi + So, Lo), Ho)

__global__ __launch_bounds__(256) void fully_fix_linear_kernel(
    const float* __restrict__ x, const float* __restrict__ w,
    const float* __restrict__ bias, float* __restrict__ out,
    int OUT, int nrows) {
  const int lane = threadIdx.x & 31;
  const int wave = threadIdx.x >> 5;
  const int row  = blockIdx.x * kWavesPerBlock + wave;   // row = b*OUT + o
  if (row >= nrows) return;                              // wave-uniform exit
  const int o = row % OUT;
  const float* xrow = x + (size_t)row * kIn;
  const float* wrow = w + (size_t)o * kIn;

#if USE_ASYNC_LDS
  __shared__ __align__(16) char lds_x[kWavesPerBlock * kRowLds];
  char* myrow = &lds_x[0] + wave * kRowLds;
  // 8 async 16B DMA copies per lane; consecutive lanes hit consecutive 16B
  // chunks in global (fully coalesced), scattered into padded LDS segments.
#pragma unroll
  for (int j = 0; j < 8; ++j) {
    const int c   = j * 32 + lane;        // chunk id 0..255 within the row
    const int seg = c >> 3, v = c & 7;    // owning lane segment / slot
    __builtin_amdgcn_global_load_async_to_lds_b128(
        (gptr_v4i)(xrow + c * 4),
        (lptr_v4i)(myrow + seg * kSegBytes + v * 16),
        0, 0);
  }
#if __has_builtin(__builtin_amdgcn_s_wait_asynccnt)
  __builtin_amdgcn_s_wait_asynccnt(0);
#else
  asm volatile("s_wait_asynccnt 0" ::: "memory");
#endif
  asm volatile("" ::: "memory");  // keep LDS reads below the wait
  const float4* x4p = (const float4*)(myrow + lane * kSegBytes);
#else
  const float4* x4p = (const float4*)xrow + lane * (kSegFloats / 4);
  __builtin_prefetch(xrow + lane * kSegFloats, 0, 0);    // global_prefetch_b8
#endif
  __builtin_prefetch(wrow + lane * kSegFloats, 0, 0);    // weight row is L2-resident

  // Build composed clamp-function for this lane's contiguous segment,
  // prepending elements from high k to low k (outer-most = lowest k).
  float S = 0.0f, Lb = -3.0e38f, Hb = 3.0e38f;
  const float4* w4p = (const float4*)wrow;
#pragma unroll
  for (int v = kSegFloats / 4 - 1; v >= 0; --v) {
    const float4 x4 = x4p[v];
    const float4 w4 = w4p[lane * (kSegFloats / 4) + v];
    const float xe[4] = {x4.x, x4.y, x4.z, x4.w};
    const float we[4] = {w4.x, w4.y, w4.z, w4.w};
#pragma unroll
    for (int e = 3; e >= 0; --e) {
      float q = floorf(xe[e] * we[e] * 32.0f);           // fix_x quantize
      q = fminf(fmaxf(q, -127.0f), 127.0f);              // fix_x clamp
      S  += q;                                            // prepend (q,-127,127)
      Lb  = fmaxf(Lb + q, -127.0f);
      Hb  = fminf(fmaxf(Hb + q, -127.0f), 127.0f);
    }
  }

  // Order-preserving wave32 butterfly: lower lane index = applied later (outer).
#pragma unroll
  for (int m = 1; m < 32; m <<= 1) {
    const float S2 = __shfl_xor(S,  m, 32);
    const float L2 = __shfl_xor(Lb, m, 32);
    const float H2 = __shfl_xor(Hb, m, 32);
    const bool lower = (lane & m) == 0;
    const float So = lower ? S  : S2, Lo = lower ? Lb : L2, Ho = lower ? Hb : H2;
    const float Si = lower ? S2 : S,  Li = lower ? L2 : Lb, Hi = lower ? H2 : Hb;
    S  = Si + So;
    Lb = fmaxf(Li + So, Lo);
    Hb = fminf(fmaxf(Hi + So, Lo), Ho);
  }
  const float acc = fminf(fmaxf(S, Lb), Hb);   // full row: f_0∘...∘f_1023 (0)

  if (lane == 0) {
    // fix_x(acc/32) is a no-op (exact 1/32 grid, in range); then fix_x(+bias):
    const float u  = acc * 0.03125f + bias[o];
    float qf = floorf(u * 32.0f);
    qf = fminf(fmaxf(qf, -127.0f), 127.0f);
    out[row] = qf * 0.03125f;
  }
}

extern "C" void kernel_launch(void* const* d_in, const int* in_sizes, int n_in,
                              void* d_out, int out_size, void* d_ws, size_t ws_size,
                              hipStream_t stream) {
  const float* x    = (const float*)d_in[0];
  const float* w    = (const float*)d_in[1];
  const float* bias = (const float*)d_in[2];
  float* out = (float*)d_out;

  const int OUT  = in_sizes[2];                 // 1024
  const int IN   = in_sizes[1] / OUT;           // 1024 (kernel assumes 1024)
  const int B    = in_sizes[0] / (OUT * IN);    // 32
  const int rows = B * OUT;                     // 32768 == out_size
  (void)IN; (void)out_size; (void)d_ws; (void)ws_size; (void)n_in;

  dim3 block(32 * kWavesPerBlock);              // 8 waves / block
  dim3 grid((rows + kWavesPerBlock - 1) / kWavesPerBlock);
  fully_fix_linear_kernel<<<grid, block, 0, stream>>>(x, w, bias, out, OUT, rows);
}